// mySAGEConv_3504693313807
// MI455X (gfx1250) — compile-verified
//
#include <hip/hip_runtime.h>

#define D       128
#define NNODES  50000
#define NEDGES  600000
#define MTILES  (NNODES / 16)   // 3125, exact

typedef float v2f __attribute__((ext_vector_type(2)));
typedef float v8f __attribute__((ext_vector_type(8)));

// ---------------------------------------------------------------------------
// Zero-init h_new (segment-max identity is 0 since all messages are >= 0).
// ---------------------------------------------------------------------------
__global__ void zero_f4_kernel(float4* __restrict__ p, int n4) {
    int i = blockIdx.x * blockDim.x + threadIdx.x;
    if (i < n4) {
        float4 z; z.x = 0.f; z.y = 0.f; z.z = 0.f; z.w = 0.f;
        p[i] = z;
    }
}

// ---------------------------------------------------------------------------
// Y[m][n] = act( sum_k X[m][k] * W[n][k] + bias0[n] (+ bias1[n]) )
// One wave per 16-row M tile; 8 WMMA accumulators cover all N=128.
// Uses V_WMMA_F32_16X16X4_F32 (exact fp32 matrix path on CDNA5).
// ---------------------------------------------------------------------------
__global__ __launch_bounds__(256) void gemm_nt_wmma_kernel(
    const float* __restrict__ X, const float* __restrict__ W,
    const float* __restrict__ bias0, const float* __restrict__ bias1,
    float* __restrict__ Y, int relu_flag)
{
    const int wid  = blockIdx.x * (blockDim.x >> 5) + (threadIdx.x >> 5);
    if (wid >= MTILES) return;                 // wave-uniform: EXEC stays all-1
    const int lane = threadIdx.x & 31;
    const int half = lane >> 4;                // 0: K=kk..kk+1, 1: K=kk+2..kk+3
    const int l15  = lane & 15;

    const float* __restrict__ xrow = X + (size_t)(wid * 16 + l15) * D;

    v8f acc[8];
#pragma unroll
    for (int n = 0; n < 8; ++n) acc[n] = (v8f){0,0,0,0,0,0,0,0};

    for (int kk = 0; kk < D; kk += 4) {
        const int kb = kk + half * 2;
        v2f a;
        a.x = xrow[kb];
        a.y = xrow[kb + 1];
#pragma unroll
        for (int n = 0; n < 8; ++n) {
            const float* __restrict__ wrow = W + (size_t)(n * 16 + l15) * D;
            v2f b;
            b.x = wrow[kb];
            b.y = wrow[kb + 1];
            // D = A(16x4 f32) * B(4x16 f32) + C
            acc[n] = __builtin_amdgcn_wmma_f32_16x16x4_f32(
                false, a, false, b, (short)0, acc[n], false, false);
        }
    }

    // C/D layout: VGPR r -> row (half*8 + r), lane l15 -> column
    const int row0 = wid * 16 + half * 8;
#pragma unroll
    for (int n = 0; n < 8; ++n) {
        const int col = n * 16 + l15;
        float badd = bias0[col];
        if (bias1) badd += bias1[col];
#pragma unroll
        for (int r = 0; r < 8; ++r) {
            float v = acc[n][r] + badd;
            if (relu_flag) v = fmaxf(v, 0.f);
            Y[(size_t)(row0 + r) * D + col] = v;
        }
    }
}

// ---------------------------------------------------------------------------
// One wave per edge: gather h_src[src[e]] (128 floats, float4 per lane),
// scale by e_weight, atomic-max into h_new[dst[e]]. All values >= 0, so
// integer atomicMax on the bit pattern == float max, and 0-init is identity.
// Working set (h_src + h_new ~= 51 MB) is resident in the 192 MB L2.
// ---------------------------------------------------------------------------
__global__ __launch_bounds__(256) void edge_scatter_max_kernel(
    const float* __restrict__ hsrc, const int* __restrict__ src,
    const int* __restrict__ dst, const float* __restrict__ ew,
    float* __restrict__ hnew)
{
    const int e = blockIdx.x * (blockDim.x >> 5) + (threadIdx.x >> 5);
    if (e >= NEDGES) return;
    const int lane = threadIdx.x & 31;

    const int   s = src[e];
    const int   d = dst[e];
    const float w = ew[e];

    float4 m = ((const float4*)(hsrc + (size_t)s * D))[lane];
    m.x *= w; m.y *= w; m.z *= w; m.w *= w;

    int* q = (int*)(hnew + (size_t)d * D) + lane * 4;
    atomicMax(q + 0, __float_as_int(m.x));
    atomicMax(q + 1, __float_as_int(m.y));
    atomicMax(q + 2, __float_as_int(m.z));
    atomicMax(q + 3, __float_as_int(m.w));
}

// ---------------------------------------------------------------------------
extern "C" void kernel_launch(void* const* d_in, const int* in_sizes, int n_in,
                              void* d_out, int out_size, void* d_ws, size_t ws_size,
                              hipStream_t stream)
{
    const float* node_feats = (const float*)d_in[0];
    const int*   src        = (const int*)  d_in[1];
    const int*   dst        = (const int*)  d_in[2];
    const float* e_weight   = (const float*)d_in[3];
    const float* pool_w     = (const float*)d_in[4];
    const float* pool_b     = (const float*)d_in[5];
    const float* lin_w      = (const float*)d_in[6];
    const float* lin_b      = (const float*)d_in[7];
    const float* bias       = (const float*)d_in[8];
    float*       out        = (float*)d_out;

    float* hsrc = (float*)d_ws;                       // 50000*128 f32 = 25.6 MB
    float* hnew = hsrc + (size_t)NNODES * D;          // +25.6 MB

    // 1) zero h_new (max identity for non-negative messages)
    const int n4 = NNODES * D / 4;
    zero_f4_kernel<<<(n4 + 255) / 256, 256, 0, stream>>>((float4*)hnew, n4);

    // 2) h_src = relu(X @ pool_w^T + pool_b)
    const int gemm_blocks = (MTILES + 7) / 8;         // 8 waves / 256-thread block
    gemm_nt_wmma_kernel<<<gemm_blocks, 256, 0, stream>>>(
        node_feats, pool_w, pool_b, nullptr, hsrc, 1);

    // 3) edge gather + scatter-max (1 wave per edge)
    const int edge_blocks = (NEDGES + 7) / 8;
    edge_scatter_max_kernel<<<edge_blocks, 256, 0, stream>>>(
        hsrc, src, dst, e_weight, hnew);

    // 4) out = h_new @ lin_w^T + lin_b + bias
    gemm_nt_wmma_kernel<<<gemm_blocks, 256, 0, stream>>>(
        hnew, lin_w, lin_b, bias, out, 0);
}